// SphericalHarmonicClebschGordanAttention_84464826843862
// MI455X (gfx1250) — compile-verified
//
#include <hip/hip_runtime.h>
#include <hip/hip_bf16.h>
#include <math.h>

typedef __attribute__((ext_vector_type(2))) float v2f;
typedef __attribute__((ext_vector_type(4))) float v4f;
typedef __attribute__((ext_vector_type(8))) float v8f;

// Problem shape is fixed by the harness; baking it in lets every load/store
// offset fold into the 24-bit instruction-immediate field (no address VALU
// per tile).
constexpr int Bc = 2, Nc = 1024, Hc = 8, Gc = 8;
constexpr int NBATCH = Bc * Hc * Gc;       // 128
constexpr int KPAD   = 16;                 // 15 SH features padded to 16

// ---------------------------------------------------------------------------
// Phase 1: rotate -> normalize -> SH(l=1,2,3) -> 15 features padded to 16.
// Input  v_in : [B, N, H, G, 3]   (q or k)
// Input  R_in : [B, N, 3, 3]
// Output Y    : [B*H*G, N, 16]    (batch-major for the GEMM; col 15 zeroed)
// ---------------------------------------------------------------------------
__global__ void sh_expand_kernel(const float* __restrict__ v_in,
                                 const float* __restrict__ R_in,
                                 float* __restrict__ Y_out,
                                 float scale)
{
    long t = (long)blockIdx.x * blockDim.x + threadIdx.x;
    const long total = (long)Bc * Nc * Hc * Gc;
    if (t >= total) return;

    int g = (int)(t % Gc); long r = t / Gc;
    int h = (int)(r % Hc); r /= Hc;
    int n = (int)(r % Nc);
    int b = (int)(r / Nc);

    const float* v = v_in + t * 3;
    const float* R = R_in + ((long)b * Nc + n) * 9;

    float vx = v[0], vy = v[1], vz = v[2];
    // q_rot_i = sum_j R[i][j] * v_j
    float x = R[0] * vx + R[1] * vy + R[2] * vz;
    float y = R[3] * vx + R[4] * vy + R[5] * vz;
    float z = R[6] * vx + R[7] * vy + R[8] * vz;

    float nrm = sqrtf(x * x + y * y + z * z);
    float inv = 1.0f / fmaxf(nrm, 1e-12f);
    x *= inv; y *= inv; z *= inv;

    float x2 = x * x, y2 = y * y, z2 = z * z;

    const float s3  = 1.7320508075688772f;   // sqrt(3)
    const float s15 = 3.8729833462074170f;   // sqrt(15)
    const float c1  = 0.6123724356957945f;   // sqrt(3/8)
    const float c3  = 0.7905694150420949f;   // sqrt(5/8)
    const float f1  = 0.4886025119029199f;   // sqrt(3)/sqrt(4pi)
    const float f2  = 0.6307831305050401f;   // sqrt(5)/sqrt(4pi)
    const float f3  = 0.7463526651802308f;   // sqrt(7)/sqrt(4pi)

    float Y[16];
    // l=1
    Y[0]  = f1 * x;
    Y[1]  = f1 * y;
    Y[2]  = f1 * z;
    // l=2
    Y[3]  = f2 * (s3 * x * z);
    Y[4]  = f2 * (s3 * x * y);
    Y[5]  = f2 * (y2 - 0.5f * (x2 + z2));
    Y[6]  = f2 * (s3 * y * z);
    Y[7]  = f2 * ((s3 * 0.5f) * (z2 - x2));
    // l=3
    Y[8]  = f3 * (c3 * x * (3.0f * z2 - x2));
    Y[9]  = f3 * (s15 * x * y * z);
    Y[10] = f3 * (c1 * x * (4.0f * y2 - x2 - z2));
    Y[11] = f3 * (0.5f * y * (2.0f * y2 - 3.0f * x2 - 3.0f * z2));
    Y[12] = f3 * (c1 * z * (4.0f * y2 - x2 - z2));
    Y[13] = f3 * ((s15 * 0.5f) * y * (z2 - x2));
    Y[14] = f3 * (c3 * z * (z2 - 3.0f * x2));
    Y[15] = 0.0f;  // K padding: WMMA K=16 step contributes nothing

    // batch-major: [(b*H + h)*G + g][n][16]
    float* o = Y_out + ((((long)b * Hc + h) * Gc + g) * Nc + n) * KPAD;
    #pragma unroll
    for (int i = 0; i < 4; ++i) {
        v4f w = { scale * Y[4 * i + 0], scale * Y[4 * i + 1],
                  scale * Y[4 * i + 2], scale * Y[4 * i + 3] };
        *(v4f*)(o + 4 * i) = w;
    }
}

// ---------------------------------------------------------------------------
// Phase 2: batched score GEMM. One wave32 computes a 16(n) x 64(m) strip:
// 4 output tiles sharing one A fragment, K=16 via 4x V_WMMA_F32_16X16X4_F32
// per tile. All B loads and C stores are immediate offsets off one base.
//
// ISA fragment layouts (wave32):
//   A 16x4 f32 : lanes 0-15 -> {K+0 (v0), K+1 (v1)}, lanes 16-31 -> {K+2, K+3}; M = lane&15
//   B 4x16 f32 : v0 -> K+0 (lanes 0-15) / K+2 (lanes 16-31); v1 -> K+1 / K+3;  N = lane&15
//   C/D 16x16  : vgpr i -> M = i + (lane>=16 ? 8 : 0), N = lane&15
// => both A and B fragments load one float2 at col (4*kk + (laneHi?2:0)).
// ---------------------------------------------------------------------------
__global__ void score_gemm_kernel(const float* __restrict__ Yq,
                                  const float* __restrict__ Yk,
                                  float* __restrict__ out)
{
    constexpr int TILES_N   = Nc >> 4;            // 64 row tiles
    constexpr int MGROUPS   = Nc >> 6;            // 16 col groups of 64
    constexpr int STRIPS_PB = TILES_N * MGROUPS;  // 1024 strips per batch

    const int lane = threadIdx.x & 31;
    const int wave = threadIdx.x >> 5;
    const int strip = blockIdx.x * 8 + wave;      // 8 waves per block

    const int batch = strip / STRIPS_PB;
    const int s     = strip % STRIPS_PB;
    const int tn    = s / MGROUPS;                // row tile (q / M)
    const int mg    = s % MGROUPS;                // 64-wide column group (k)

    const int hi  = lane >> 4;                    // 0: K base +0, 1: K base +2
    const int l16 = lane & 15;

    const float* Arow = Yq + ((long)batch * Nc + tn * 16 + l16) * KPAD + hi * 2;
    const float* Brow = Yk + ((long)batch * Nc + mg * 64 + l16) * KPAD + hi * 2;

    // A fragments for all 4 K-steps (row is contiguous: cols {0..15})
    v2f a[4];
    #pragma unroll
    for (int kk = 0; kk < 4; ++kk)
        a[kk] = *(const v2f*)(Arow + 4 * kk);

    // B fragments for 4 tiles x 4 K-steps; tile t is +256 floats (immediate)
    v2f bf[4][4];
    #pragma unroll
    for (int tile = 0; tile < 4; ++tile)
        #pragma unroll
        for (int kk = 0; kk < 4; ++kk)
            bf[tile][kk] = *(const v2f*)(Brow + tile * (16 * KPAD) + 4 * kk);

    v8f c[4] = {v8f{}, v8f{}, v8f{}, v8f{}};
    // kk outer / tile inner: 4 independent accumulator chains interleaved so
    // consecutive WMMAs never depend on each other's D.
    #pragma unroll
    for (int kk = 0; kk < 4; ++kk)
        #pragma unroll
        for (int tile = 0; tile < 4; ++tile)
            c[tile] = __builtin_amdgcn_wmma_f32_16x16x4_f32(
                          false, a[kk], false, bf[tile][kk],
                          (short)0, c[tile], false, false);

    // D[vgpr i] of tile t -> row (tn*16 + i + hi*8), col (mg*64 + t*16 + l16)
    float* obase = out + ((long)batch * Nc + tn * 16 + hi * 8) * (long)Nc
                       + mg * 64 + l16;
    #pragma unroll
    for (int i = 0; i < 8; ++i)
        #pragma unroll
        for (int tile = 0; tile < 4; ++tile)
            obase[i * Nc + tile * 16] = c[tile][i];   // all-immediate offsets
}

// ---------------------------------------------------------------------------
// Host launcher
// ---------------------------------------------------------------------------
extern "C" void kernel_launch(void* const* d_in, const int* in_sizes, int n_in,
                              void* d_out, int out_size, void* d_ws, size_t ws_size,
                              hipStream_t stream)
{
    const float* q   = (const float*)d_in[0];  // [B,N,H,G,3]
    const float* k   = (const float*)d_in[1];  // [B,N,H,G,3]
    const float* r_q = (const float*)d_in[2];  // [B,N,3,3]
    const float* r_k = (const float*)d_in[3];  // [B,N,3,3]
    float* out = (float*)d_out;                // [B,H,G,N,N]

    float* Yq = (float*)d_ws;                            // [128,1024,16] f32
    float* Yk = Yq + (size_t)NBATCH * Nc * KPAD;         // +8.39 MB

    const long total   = (long)Bc * Nc * Hc * Gc;        // 131072
    const int  blocks1 = (int)((total + 255) / 256);

    const float inv_sqrt15 = 0.25819888974716115f;       // 1/sqrt(15)
    sh_expand_kernel<<<blocks1, 256, 0, stream>>>(q, r_q, Yq, inv_sqrt15);
    sh_expand_kernel<<<blocks1, 256, 0, stream>>>(k, r_k, Yk, 1.0f);

    // One wave per 16x64 strip: 128 batches * 64 * 16 = 131072 strips,
    // 8 waves (256 threads) per block.
    const int totalStrips = NBATCH * (Nc / 16) * (Nc / 64);
    const int blocks2     = totalStrips / 8;             // 16384
    score_gemm_kernel<<<blocks2, 256, 0, stream>>>(Yq, Yk, out);
}